// KNNSegmentator_39281770889915
// MI455X (gfx1250) — compile-verified
//
#include <hip/hip_runtime.h>
#include <hip/hip_bf16.h>

// ---------------- problem constants ----------------
constexpr int BS      = 16;
constexpr int IMG     = 224;
constexpr int PATCH   = 16;
constexpr int NROWS   = 14;     // IMG/PATCH
constexpr int P       = 196;    // NROWS*NROWS
constexpr int P2      = 256;    // PATCH*PATCH
constexpr int D       = 768;
constexpr int NTRAIN  = 2000;
constexpr int K       = 20;
constexpr int C       = 21;
constexpr int NTILES  = NTRAIN / 16;  // 125

typedef __attribute__((ext_vector_type(2))) float v2f;
typedef __attribute__((ext_vector_type(8))) float v8f;

// =====================================================================
// Kernel 1: per-patch GEMM  sim[b,p,n] = sum_d test[b,p,d]*train[p,d,n]
// One wave (32 lanes) per (n-tile, p): M=16 (batch), N=16, K=768 by 4.
// Uses V_WMMA_F32_16X16X4_F32. Memory-bound on train_features stream.
// =====================================================================
__global__ __launch_bounds__(32) void knn_sim_gemm(
    const float* __restrict__ test,     // [BS, P, D]
    const float* __restrict__ train,    // [P, D, NTRAIN]
    float* __restrict__ sim)            // [BS, P, NTRAIN]
{
    const int ntile = blockIdx.x;            // 0..124
    const int p     = blockIdx.y;            // 0..195
    const int lane  = threadIdx.x;           // 0..31
    const int n0    = ntile * 16;
    const int half  = lane >> 4;             // 0 | 1
    const int l16   = lane & 15;
    const int kbase = half * 2;              // K pair owned by this half-wave

    // A fragment source: A[M=l16][K] = test[l16, p, k+K]
    const float* aptr = test + ((size_t)l16 * P + p) * D + kbase;
    // B fragment source: B[K][N=l16] = train[p, k+K, n0+l16]
    const float* bptr = train + ((size_t)p * D + kbase) * NTRAIN + n0 + l16;

    v8f acc = {};
#pragma unroll 8
    for (int k = 0; k < D; k += 4) {
        v2f a, b;
        a.x = aptr[k];
        a.y = aptr[k + 1];
        b.x = bptr[(size_t)k * NTRAIN];
        b.y = bptr[(size_t)(k + 1) * NTRAIN];
        // (neg_a, A, neg_b, B, c_mod, C, reuse_a, reuse_b)
        acc = __builtin_amdgcn_wmma_f32_16x16x4_f32(
            false, a, false, b, (short)0, acc, false, false);
    }

    // C/D layout: VGPR r -> M = r + 8*half, N = l16
#pragma unroll
    for (int r = 0; r < 8; ++r) {
        const int m = r + half * 8;
        sim[((size_t)m * P + p) * NTRAIN + n0 + l16] = acc[r];
    }
}

// =====================================================================
// Kernel 2: top-K (K=20) over N=2000 per row, ties -> smallest index.
// One 256-thread block per (b,p) row; sim row staged in LDS.
// =====================================================================
__global__ __launch_bounds__(256) void knn_topk(
    const float* __restrict__ sim,   // [BS*P, NTRAIN]
    float* __restrict__ top_sim,     // [BS*P, K]
    int*   __restrict__ top_idx)     // [BS*P, K]
{
    __shared__ float sv[NTRAIN];
    __shared__ float rv[256];
    __shared__ int   ri[256];

    const int row = blockIdx.x;            // b*P + p
    const int t   = threadIdx.x;
    const float* src = sim + (size_t)row * NTRAIN;

    for (int i = t; i < NTRAIN; i += 256) sv[i] = src[i];
    __syncthreads();

    const float NEG = -3.402823466e38f;
    for (int k = 0; k < K; ++k) {
        float best = NEG;
        int   bi   = 0x7fffffff;
        for (int i = t; i < NTRAIN; i += 256) {
            float v = sv[i];
            if (v > best || (v == best && i < bi)) { best = v; bi = i; }
        }
        rv[t] = best; ri[t] = bi;
        __syncthreads();
        for (int s = 128; s > 0; s >>= 1) {
            if (t < s) {
                float ov = rv[t + s]; int oi = ri[t + s];
                if (ov > rv[t] || (ov == rv[t] && oi < ri[t])) {
                    rv[t] = ov; ri[t] = oi;
                }
            }
            __syncthreads();
        }
        if (t == 0) {
            top_sim[(size_t)row * K + k] = rv[0];
            top_idx[(size_t)row * K + k] = ri[0];
            sv[ri[0]] = NEG;               // mask out for next round
        }
        __syncthreads();
    }
}

// =====================================================================
// Kernel 3: softmax over K (constant per patch), gather per-pixel
// labels of the K neighbors, weighted vote over C=21, argmax, scatter
// into make_grid image layout. One block per (b,p), thread = pixel.
// =====================================================================
__global__ __launch_bounds__(256) void knn_vote(
    const float* __restrict__ top_sim,   // [BS*P, K]
    const int*   __restrict__ top_idx,   // [BS*P, K]
    const int*   __restrict__ labels,    // [P, P2, NTRAIN]
    float*       __restrict__ out)       // [BS, IMG, IMG] as float
{
    __shared__ float w[K];
    __shared__ int   idx[K];
    __shared__ float vote[256 * C];      // [C][256] class-major, lane = px -> bank-conflict free

    const int bp = blockIdx.x;           // b*P + p
    const int b  = bp / P;
    const int p  = bp % P;
    const int px = threadIdx.x;          // pixel in patch, 0..255

    if (px < K) {
        w[px]   = top_sim[(size_t)bp * K + px];
        idx[px] = top_idx[(size_t)bp * K + px];
    }
    __syncthreads();

    if (px == 0) {                        // softmax over the K sims (tiny)
        float m = w[0];
        for (int k = 1; k < K; ++k) m = fmaxf(m, w[k]);
        float s = 0.f;
        for (int k = 0; k < K; ++k) { w[k] = __expf(w[k] - m); s += w[k]; }
        float inv = 1.f / s;
        for (int k = 0; k < K; ++k) w[k] *= inv;
    }
    __syncthreads();

    float* v = &vote[px];                 // strided per-thread vote bins
#pragma unroll
    for (int c = 0; c < C; ++c) v[(size_t)c * 256] = 0.f;

    const int* lab = labels + ((size_t)p * P2 + px) * NTRAIN;
#pragma unroll
    for (int k = 0; k < K; ++k) {
        int cls = lab[idx[k]];
        v[(size_t)cls * 256] += w[k];
    }

    int   best = 0;
    float bv   = v[0];
#pragma unroll
    for (int c = 1; c < C; ++c) {
        float vv = v[(size_t)c * 256];
        if (vv > bv) { bv = vv; best = c; } // ties keep first index (jnp.argmax)
    }

    const int gr = p / NROWS, gc = p % NROWS;
    const int pr = px / PATCH, pc = px % PATCH;
    const int y  = gr * PATCH + pr;
    const int x  = gc * PATCH + pc;
    out[((size_t)b * IMG + y) * IMG + x] = (float)best;
}

// =====================================================================
extern "C" void kernel_launch(void* const* d_in, const int* in_sizes, int n_in,
                              void* d_out, int out_size, void* d_ws, size_t ws_size,
                              hipStream_t stream) {
    const float* test   = (const float*)d_in[0];   // [BS, P, D]
    const float* train  = (const float*)d_in[1];   // [P, D, NTRAIN]
    const int*   labels = (const int*)d_in[2];     // [P, P2, NTRAIN]
    float*       out    = (float*)d_out;           // [BS, IMG, IMG]

    // workspace layout
    float* sim  = (float*)d_ws;                             // BS*P*NTRAIN
    float* tsim = sim + (size_t)BS * P * NTRAIN;            // BS*P*K
    int*   tidx = (int*)(tsim + (size_t)BS * P * K);        // BS*P*K

    dim3 g1(NTILES, P);
    knn_sim_gemm<<<g1, 32, 0, stream>>>(test, train, sim);

    knn_topk<<<BS * P, 256, 0, stream>>>(sim, tsim, tidx);

    knn_vote<<<BS * P, 256, 0, stream>>>(tsim, tidx, labels, out);
}